// GCN_36816459661702
// MI455X (gfx1250) — compile-verified
//
#include <hip/hip_runtime.h>
#include <math.h>

// ---- problem constants (match reference) ----
#define N_NODES 50000
#define N_EDGES 800000
#define F_IN    128
#define HIDDEN  96
#define N_CLASS 40

typedef __attribute__((ext_vector_type(2))) float v2f;
typedef __attribute__((ext_vector_type(8))) float v8f;

__device__ __forceinline__ float elu1(float x) {
    return x > 0.f ? x : (expf(x) - 1.f);
}

// ---------------- small elementwise / edge kernels ----------------

__global__ void k_fill(float* __restrict__ p, float v, int n) {
    int i = blockIdx.x * blockDim.x + threadIdx.x;
    if (i < n) p[i] = v;
}

// deg[c] += w[e]  (deg pre-filled with 1.0 for the self-loop weight)
__global__ void k_deg(const int* __restrict__ cols, const float* __restrict__ w,
                      float* __restrict__ deg, int E) {
    int e = blockIdx.x * blockDim.x + threadIdx.x;
    if (e < E) atomicAdd(&deg[cols[e]], w[e]);
}

// in-place: d[i] = d[i] > 0 ? rsqrt(d[i]) : 0
__global__ void k_dinv(float* __restrict__ d, int n) {
    int i = blockIdx.x * blockDim.x + threadIdx.x;
    if (i < n) {
        float v = d[i];
        d[i] = v > 0.f ? rsqrtf(v) : 0.f;
    }
}

// norm[e] = dinv[row]*w*dinv[col]
__global__ void k_norm(const int* __restrict__ rows, const int* __restrict__ cols,
                       const float* __restrict__ w, const float* __restrict__ dinv,
                       float* __restrict__ norm, int E) {
    int e = blockIdx.x * blockDim.x + threadIdx.x;
    if (e < E) norm[e] = dinv[rows[e]] * w[e] * dinv[cols[e]];
}

// self-loop contribution: out[i,:] = dinv[i]^2 * h[i,:]   (float4-vectorized)
__global__ void k_selfinit(const float* __restrict__ h, const float* __restrict__ dinv,
                           float* __restrict__ out) {
    int idx = blockIdx.x * blockDim.x + threadIdx.x;       // over N_NODES * HIDDEN/4
    const int per_row = HIDDEN / 4;
    if (idx >= N_NODES * per_row) return;
    int i = idx / per_row;
    float d = dinv[i];
    float d2 = d * d;
    float4 hv = ((const float4*)h)[idx];
    float4 o;
    o.x = d2 * hv.x; o.y = d2 * hv.y; o.z = d2 * hv.z; o.w = d2 * hv.w;
    ((float4*)out)[idx] = o;
}

// one wave (32 lanes) per edge; each lane does 3 feature atomics (96 = 3*32)
__global__ __launch_bounds__(256) void k_scatter(const int* __restrict__ rows,
                                                 const int* __restrict__ cols,
                                                 const float* __restrict__ norm,
                                                 const float* __restrict__ h,
                                                 float* __restrict__ out) {
    int wave = threadIdx.x >> 5;
    int lane = threadIdx.x & 31;
    int e = blockIdx.x * 8 + wave;
    if (e >= N_EDGES) return;
    int r = rows[e];
    int c = cols[e];
    float w = norm[e];
    const float* hr = h + (size_t)r * HIDDEN;
    float* oc = out + (size_t)c * HIDDEN;
#pragma unroll
    for (int j = 0; j < 3; ++j) {
        int f = lane + 32 * j;
        atomicAdd(oc + f, w * hr[f]);
    }
}

// out[idx] = elu(prop[idx] + b[idx % HIDDEN])
__global__ void k_bias_elu(const float* __restrict__ prop, const float* __restrict__ b,
                           float* __restrict__ out, int total) {
    int idx = blockIdx.x * blockDim.x + threadIdx.x;
    if (idx < total) out[idx] = elu1(prop[idx] + b[idx % HIDDEN]);
}

// out[idx] = res[idx] + elu(prop[idx] + b[idx % HIDDEN])
__global__ void k_bias_elu_res(const float* __restrict__ prop, const float* __restrict__ b,
                               const float* __restrict__ res, float* __restrict__ out,
                               int total) {
    int idx = blockIdx.x * blockDim.x + threadIdx.x;
    if (idx < total) out[idx] = res[idx] + elu1(prop[idx] + b[idx % HIDDEN]);
}

// in-place log-softmax over N_CLASS=40 cols; one wave per node
__global__ __launch_bounds__(256) void k_logsoftmax(float* __restrict__ L) {
    int wave = threadIdx.x >> 5;
    int lane = threadIdx.x & 31;
    int i = blockIdx.x * 8 + wave;
    if (i >= N_NODES) return;
    float* row = L + (size_t)i * N_CLASS;
    float v0 = row[lane];                                      // lane < 32 < 40: valid
    bool hi = (lane + 32) < N_CLASS;
    float v1 = hi ? row[lane + 32] : -INFINITY;
    float m = fmaxf(v0, v1);
#pragma unroll
    for (int off = 16; off > 0; off >>= 1) m = fmaxf(m, __shfl_xor(m, off, 32));
    float s = expf(v0 - m) + (hi ? expf(v1 - m) : 0.f);
#pragma unroll
    for (int off = 16; off > 0; off >>= 1) s += __shfl_xor(s, off, 32);
    float lse = m + logf(s);
    row[lane] = v0 - lse;
    if (hi) row[lane + 32] = v1 - lse;
}

// ---------------- fp32 WMMA GEMM: C[M x NOUT] = A[M x K] * W[K x NOUT] ----------------
// Block = 256 threads = 8 wave32 waves; wave w owns M-tile (blockIdx*8 + w) of 16 rows.
// W staged in LDS, zero-padded to NPAD columns, in k-pair-interleaved layout:
//   sW[((k>>1)*NPAD + n)*2 + (k&1)]
// so the two K-values a lane consumes (k0+2*half, k0+2*half+1) are one aligned
// 8-byte ds_load_b64 landing directly in a WMMA-ready even VGPR pair.
// Fragment layouts per CDNA5 ISA 7.12.2 (f32 16x16x4):
//   A: lanes 0-15 v0/v1 = A[m][k0],A[m][k0+1]; lanes 16-31 = A[m][k0+2],A[m][k0+3]
//   B: v0 = W[k0 (+2 for hi half)][n]; v1 = W[k0+1 (+2)][n]
//   C/D: vgpr r = row m0+r (lanes 0-15) / m0+r+8 (lanes 16-31), col = lane&15 (+16*t)
template <int K, int NPAD, int NOUT, bool BIAS, bool ELU_>
__global__ __launch_bounds__(256) void k_gemm_wmma(const float* __restrict__ A,
                                                   const float* __restrict__ W,
                                                   const float* __restrict__ bias,
                                                   float* __restrict__ C, int M) {
    constexpr int NT = NPAD / 16;
    __shared__ float sW[K * NPAD];
    int tid = threadIdx.x;
    for (int idx = tid; idx < K * NPAD; idx += 256) {
        int k = idx / NPAD;
        int n = idx % NPAD;
        int dst = (((k >> 1) * NPAD) + n) * 2 + (k & 1);
        sW[dst] = (n < NOUT) ? W[k * NOUT + n] : 0.f;
    }
    __syncthreads();

    int wave = tid >> 5;
    int lane = tid & 31;
    int mt = blockIdx.x * 8 + wave;
    if (mt * 16 >= M) return;        // wave-uniform: EXEC stays all-1s for WMMA
    int m0 = mt * 16;
    int half = lane >> 4;            // 0: K-lo pair, 1: K-hi pair
    int lm = lane & 15;

    const float* Arow = A + (size_t)(m0 + lm) * K + 2 * half;
    // per-lane base into the pair-interleaved B: pair p = k0/2 + half, col = lm
    const float* Bbase = &sW[(half * NPAD + lm) * 2];

    v8f acc[NT];
    v8f z = {0.f, 0.f, 0.f, 0.f, 0.f, 0.f, 0.f, 0.f};
#pragma unroll
    for (int t = 0; t < NT; ++t) acc[t] = z;

    for (int k0 = 0; k0 < K; k0 += 4) {
        float2 a2 = *(const float2*)(Arow + k0);   // 8B aligned: k0 even, row stride even
        v2f av;
        av.x = a2.x;
        av.y = a2.y;
        const float* bp = Bbase + (k0 >> 1) * (NPAD * 2);
#pragma unroll
        for (int t = 0; t < NT; ++t) {
            v2f bv = *(const v2f*)(bp + t * 32);   // single aligned ds_load_b64
            acc[t] = __builtin_amdgcn_wmma_f32_16x16x4_f32(
                false, av, false, bv, (short)0, acc[t], false, false);
        }
    }

#pragma unroll
    for (int t = 0; t < NT; ++t) {
        int col = t * 16 + lm;
        if (col < NOUT) {
#pragma unroll
            for (int r = 0; r < 8; ++r) {
                int row = m0 + r + 8 * half;
                float v = acc[t][r];
                if (BIAS) v += bias[col];
                if (ELU_) v = elu1(v);
                C[(size_t)row * NOUT + col] = v;
            }
        }
    }
}

// ---------------- host-side orchestration ----------------

extern "C" void kernel_launch(void* const* d_in, const int* in_sizes, int n_in,
                              void* d_out, int out_size, void* d_ws, size_t ws_size,
                              hipStream_t stream) {
    (void)in_sizes; (void)n_in; (void)out_size; (void)ws_size;

    const float* x   = (const float*)d_in[0];
    const int*   ei  = (const int*)d_in[1];     // [2, E]: rows then cols
    const float* ew  = (const float*)d_in[2];
    const float* Wc1 = (const float*)d_in[3];
    const float* bc1 = (const float*)d_in[4];
    const float* Wc2 = (const float*)d_in[5];
    const float* bc2 = (const float*)d_in[6];
    const float* Wc3 = (const float*)d_in[7];
    const float* bc3 = (const float*)d_in[8];
    const float* Wl1 = (const float*)d_in[9];
    const float* bl1 = (const float*)d_in[10];
    const float* Wl2 = (const float*)d_in[11];
    const float* bl2 = (const float*)d_in[12];
    float* out = (float*)d_out;

    const int* rows = ei;
    const int* cols = ei + N_EDGES;

    // workspace carve-up (floats), 256B-aligned chunks
    float* wsf = (float*)d_ws;
    size_t off = 0;
    auto carve = [&](size_t n) {
        float* p = wsf + off;
        off += (n + 63) & ~(size_t)63;
        return p;
    };
    float* dinv = carve(N_NODES);                       // deg, then deg^-1/2 in place
    float* norm = carve(N_EDGES);
    float* htmp = carve((size_t)N_NODES * HIDDEN);      // GEMM output h = x@W
    float* prop = carve((size_t)N_NODES * HIDDEN);      // scatter accumulator
    float* X1   = carve((size_t)N_NODES * HIDDEN);      // elu(conv1(x)) residual
    float* Hc   = carve((size_t)N_NODES * HIDDEN);      // current hidden state

    const int TPB = 256;
    const int gN    = (N_NODES + TPB - 1) / TPB;                  // 196
    const int gE    = (N_EDGES + TPB - 1) / TPB;                  // 3125
    const int gNH4  = (N_NODES * (HIDDEN / 4) + TPB - 1) / TPB;   // 4688
    const int gNH   = (N_NODES * HIDDEN + TPB - 1) / TPB;         // 18750
    const int gScat = (N_EDGES + 7) / 8;                          // 100000
    const int gGemm = ((N_NODES + 15) / 16 + 7) / 8;              // 391
    const int gSmax = (N_NODES + 7) / 8;                          // 6250
    const int NHtot = N_NODES * HIDDEN;

    // ---- gcn_norm ----
    k_fill<<<gN, TPB, 0, stream>>>(dinv, 1.0f, N_NODES);          // self-loop weight
    k_deg<<<gE, TPB, 0, stream>>>(cols, ew, dinv, N_EDGES);
    k_dinv<<<gN, TPB, 0, stream>>>(dinv, N_NODES);
    k_norm<<<gE, TPB, 0, stream>>>(rows, cols, ew, dinv, norm, N_EDGES);

    // ---- conv1: X1 = elu(propagate(x @ Wc1) + bc1) ----
    k_gemm_wmma<F_IN, HIDDEN, HIDDEN, false, false>
        <<<gGemm, TPB, 0, stream>>>(x, Wc1, nullptr, htmp, N_NODES);
    k_selfinit<<<gNH4, TPB, 0, stream>>>(htmp, dinv, prop);
    k_scatter<<<gScat, TPB, 0, stream>>>(rows, cols, norm, htmp, prop);
    k_bias_elu<<<gNH, TPB, 0, stream>>>(prop, bc1, X1, NHtot);

    // ---- Hc = elu(x @ Wl1 + bl1) ----
    k_gemm_wmma<F_IN, HIDDEN, HIDDEN, true, true>
        <<<gGemm, TPB, 0, stream>>>(x, Wl1, bl1, Hc, N_NODES);

    // ---- conv2: Hc = elu(propagate(Hc @ Wc2) + bc2) ----
    k_gemm_wmma<HIDDEN, HIDDEN, HIDDEN, false, false>
        <<<gGemm, TPB, 0, stream>>>(Hc, Wc2, nullptr, htmp, N_NODES);
    k_selfinit<<<gNH4, TPB, 0, stream>>>(htmp, dinv, prop);
    k_scatter<<<gScat, TPB, 0, stream>>>(rows, cols, norm, htmp, prop);
    k_bias_elu<<<gNH, TPB, 0, stream>>>(prop, bc2, Hc, NHtot);

    // ---- conv3 + residual: Hc = X1 + elu(propagate(Hc @ Wc3) + bc3) ----
    k_gemm_wmma<HIDDEN, HIDDEN, HIDDEN, false, false>
        <<<gGemm, TPB, 0, stream>>>(Hc, Wc3, nullptr, htmp, N_NODES);
    k_selfinit<<<gNH4, TPB, 0, stream>>>(htmp, dinv, prop);
    k_scatter<<<gScat, TPB, 0, stream>>>(rows, cols, norm, htmp, prop);
    k_bias_elu_res<<<gNH, TPB, 0, stream>>>(prop, bc3, X1, Hc, NHtot);

    // ---- logits = Hc @ Wl2 + bl2 ; log_softmax in place ----
    k_gemm_wmma<HIDDEN, 48, N_CLASS, true, false>
        <<<gGemm, TPB, 0, stream>>>(Hc, Wl2, bl2, out, N_NODES);
    k_logsoftmax<<<gSmax, TPB, 0, stream>>>(out);
}